// GATConv_34617436406126
// MI455X (gfx1250) — compile-verified
//
#include <hip/hip_runtime.h>
#include <hip/hip_bf16.h>

// ---------------------------------------------------------------------------
// GATConv forward for MI455X (gfx1250, wave32).
//   N=100000 nodes, E=1600000 edges, D_NODE=256, D_EDGE=16, H=4, O=32.
// Dense projections use V_WMMA_F32_16X16X4_F32 (fp32-exact matrix path).
// Edge phase uses float atomics against L2-resident (51MB < 192MB) buffers.
// ---------------------------------------------------------------------------

typedef __attribute__((ext_vector_type(2))) float v2f;
typedef __attribute__((ext_vector_type(8))) float v8f;

#define DNODE   256
#define HO      128            // HEADS*OUT
#define HEADS   4
#define NEG_SLOPE 0.2f
#define LDSP    257            // padded LDS row stride (floats) -> conflict-free

// order-preserving float <-> uint bijection for atomic max on floats
__device__ __forceinline__ unsigned f2ord(float f) {
    unsigned u = __float_as_uint(f);
    return (u & 0x80000000u) ? ~u : (u | 0x80000000u);
}
__device__ __forceinline__ float ord2f(unsigned u) {
    unsigned v = (u & 0x80000000u) ? (u & 0x7FFFFFFFu) : ~u;
    return __uint_as_float(v);
}

// ---------------------------------------------------------------------------
// Kernel 1: node projections via WMMA.
//   feat_src = X @ Wsrc^T          -> ws
//   out      = X @ Wdst^T + b_dst  -> d_out (residual base; aggregation adds)
// One block = one 16-node tile; 8 waves each own a 16-wide column tile.
// ---------------------------------------------------------------------------
__global__ __launch_bounds__(256) void gat_node_proj(
    const float* __restrict__ X,
    const float* __restrict__ Wsrc,
    const float* __restrict__ Wdst,
    const float* __restrict__ bdst,
    float* __restrict__ feat_src,
    float* __restrict__ out,
    int N)
{
    __shared__ float sA[16 * LDSP];

    const int i0 = blockIdx.x * 16;
    const int t  = threadIdx.x;

    // cooperative, coalesced tile load: 16 rows x 256 cols
    #pragma unroll
    for (int j = 0; j < 16; ++j) {
        float v = 0.0f;
        if (i0 + j < N) v = X[(size_t)(i0 + j) * DNODE + t];
        sA[j * LDSP + t] = v;
    }
    __syncthreads();

    const int wave  = t >> 5;        // 0..7 -> column tile
    const int lane  = t & 31;
    const int n0    = wave * 16;     // output column base (0..112)
    const int mrow  = lane & 15;     // A: M index / B,C,D: N index
    const int khalf = (lane >> 4) * 2;  // 0 for lanes 0-15, 2 for lanes 16-31

    v8f cs = {};   // feat_src accumulator
    v8f cd = {};   // feat_dst accumulator

    const float* __restrict__ wsrow = Wsrc + (size_t)(n0 + mrow) * DNODE;
    const float* __restrict__ wdrow = Wdst + (size_t)(n0 + mrow) * DNODE;
    const float* __restrict__ arow  = sA + mrow * LDSP;

    for (int k = 0; k < DNODE; k += 4) {
        // A 16x4 f32: VGPR0 = K {k | k+2}, VGPR1 = K {k+1 | k+3}
        v2f a;
        a.x = arow[k + khalf];
        a.y = arow[k + khalf + 1];
        // B 4x16 f32 (B[k][n] = W[n][k]): same lane-half K split as A
        v2f bs, bd;
        bs.x = wsrow[k + khalf];
        bs.y = wsrow[k + khalf + 1];
        bd.x = wdrow[k + khalf];
        bd.y = wdrow[k + khalf + 1];
        cs = __builtin_amdgcn_wmma_f32_16x16x4_f32(false, a, false, bs,
                                                   (short)0, cs, false, false);
        cd = __builtin_amdgcn_wmma_f32_16x16x4_f32(false, a, false, bd,
                                                   (short)0, cd, false, false);
    }

    // C/D layout: VGPR r holds M = r (lanes 0-15) or M = r+8 (lanes 16-31),
    // N = lane%16.
    const int colg  = n0 + mrow;
    const float bias = bdst[colg];
    const int rbase = i0 + ((lane >> 4) << 3);
    #pragma unroll
    for (int r = 0; r < 8; ++r) {
        const int row = rbase + r;
        if (row < N) {
            feat_src[(size_t)row * HO + colg] = cs[r];
            out[(size_t)row * HO + colg]      = cd[r] + bias;
        }
    }
}

// ---------------------------------------------------------------------------
// Kernel 2: attention scalars a_src = X@Wa_src^T, a_dst = X@Wa_dst^T.
// One wave per node; 8-deep per-lane partials + wave32 butterfly reduction.
// ---------------------------------------------------------------------------
__global__ __launch_bounds__(256) void gat_node_attn(
    const float* __restrict__ X,
    const float* __restrict__ WaS,
    const float* __restrict__ WaD,
    float* __restrict__ a_src,
    float* __restrict__ a_dst,
    int N)
{
    const int wave = threadIdx.x >> 5;
    const int lane = threadIdx.x & 31;
    const int n = blockIdx.x * 8 + wave;
    if (n >= N) return;

    float x[8];
    #pragma unroll
    for (int j = 0; j < 8; ++j)
        x[j] = X[(size_t)n * DNODE + j * 32 + lane];

    float s0, s1, s2, s3, d0, d1, d2, d3;
    s0 = s1 = s2 = s3 = d0 = d1 = d2 = d3 = 0.0f;
    #pragma unroll
    for (int j = 0; j < 8; ++j) {
        const int c = j * 32 + lane;
        const float xv = x[j];
        s0 += xv * WaS[0 * DNODE + c];
        s1 += xv * WaS[1 * DNODE + c];
        s2 += xv * WaS[2 * DNODE + c];
        s3 += xv * WaS[3 * DNODE + c];
        d0 += xv * WaD[0 * DNODE + c];
        d1 += xv * WaD[1 * DNODE + c];
        d2 += xv * WaD[2 * DNODE + c];
        d3 += xv * WaD[3 * DNODE + c];
    }
    #pragma unroll
    for (int m = 16; m > 0; m >>= 1) {
        s0 += __shfl_xor(s0, m, 32);
        s1 += __shfl_xor(s1, m, 32);
        s2 += __shfl_xor(s2, m, 32);
        s3 += __shfl_xor(s3, m, 32);
        d0 += __shfl_xor(d0, m, 32);
        d1 += __shfl_xor(d1, m, 32);
        d2 += __shfl_xor(d2, m, 32);
        d3 += __shfl_xor(d3, m, 32);
    }
    if (lane == 0) {
        a_src[n * 4 + 0] = s0; a_src[n * 4 + 1] = s1;
        a_src[n * 4 + 2] = s2; a_src[n * 4 + 3] = s3;
        a_dst[n * 4 + 0] = d0; a_dst[n * 4 + 1] = d1;
        a_dst[n * 4 + 2] = d2; a_dst[n * 4 + 3] = d3;
    }
}

// ---------------------------------------------------------------------------
// Kernel 3: per-edge logits e = leaky(a_src[src] + a_dst[dst] + edge@Wa_e^T),
// store e, and segment max via ordered-uint atomicMax.
// ---------------------------------------------------------------------------
__global__ __launch_bounds__(256) void gat_edge_logits(
    const float* __restrict__ Ein,
    const int* __restrict__ src,
    const int* __restrict__ dst,
    const float* __restrict__ WaE,
    const float* __restrict__ a_src,
    const float* __restrict__ a_dst,
    float* __restrict__ ebuf,
    unsigned* __restrict__ emax_u,
    int E)
{
    const int e = blockIdx.x * blockDim.x + threadIdx.x;
    if (e >= E) return;
    const int s = src[e];
    const int d = dst[e];

    const float4* ein = (const float4*)(Ein + (size_t)e * 16);
    const float4 x0 = ein[0], x1 = ein[1], x2 = ein[2], x3 = ein[3];

    float eh[4];
    #pragma unroll
    for (int h = 0; h < HEADS; ++h) {
        const float* __restrict__ w = WaE + h * 16;  // uniform -> scalar loads
        float dotv =
            x0.x * w[0]  + x0.y * w[1]  + x0.z * w[2]  + x0.w * w[3]  +
            x1.x * w[4]  + x1.y * w[5]  + x1.z * w[6]  + x1.w * w[7]  +
            x2.x * w[8]  + x2.y * w[9]  + x2.z * w[10] + x2.w * w[11] +
            x3.x * w[12] + x3.y * w[13] + x3.z * w[14] + x3.w * w[15];
        float v = a_src[s * 4 + h] + a_dst[d * 4 + h] + dotv;
        v = (v > 0.0f) ? v : NEG_SLOPE * v;
        eh[h] = v;
        atomicMax(emax_u + d * 4 + h, f2ord(v));
    }
    float4 st = make_float4(eh[0], eh[1], eh[2], eh[3]);
    *(float4*)(ebuf + (size_t)e * 4) = st;
}

// ---------------------------------------------------------------------------
// Kernel 4: ee = exp(e - emax[dst]) in-place; denom[dst] += ee (atomic).
// ---------------------------------------------------------------------------
__global__ __launch_bounds__(256) void gat_softmax_norm(
    const int* __restrict__ dst,
    float* __restrict__ ebuf,
    const unsigned* __restrict__ emax_u,
    float* __restrict__ denom,
    int E)
{
    const int e = blockIdx.x * blockDim.x + threadIdx.x;
    if (e >= E) return;
    const int d = dst[e];
    float4 ev = *(float4*)(ebuf + (size_t)e * 4);
    const float e0 = expf(ev.x - ord2f(emax_u[d * 4 + 0]));
    const float e1 = expf(ev.y - ord2f(emax_u[d * 4 + 1]));
    const float e2 = expf(ev.z - ord2f(emax_u[d * 4 + 2]));
    const float e3 = expf(ev.w - ord2f(emax_u[d * 4 + 3]));
    atomicAdd(denom + d * 4 + 0, e0);
    atomicAdd(denom + d * 4 + 1, e1);
    atomicAdd(denom + d * 4 + 2, e2);
    atomicAdd(denom + d * 4 + 3, e3);
    *(float4*)(ebuf + (size_t)e * 4) = make_float4(e0, e1, e2, e3);
}

// ---------------------------------------------------------------------------
// Kernel 5: out[dst] += feat_src[src] * (ee / denom[dst]).
// Thread per (edge, channel); feat_src & out live in L2 (51MB each < 192MB).
// ---------------------------------------------------------------------------
__global__ __launch_bounds__(256) void gat_aggregate(
    const int* __restrict__ src,
    const int* __restrict__ dst,
    const float* __restrict__ feat_src,
    const float* __restrict__ ebuf,
    const float* __restrict__ denom,
    float* __restrict__ out,
    int E)
{
    const int gid = blockIdx.x * 256 + threadIdx.x;   // E*128 = 204.8M < 2^31
    const int e = gid >> 7;
    const int c = gid & 127;
    if (e >= E) return;
    const int h = c >> 5;
    const int s = src[e];
    const int d = dst[e];
    const float attn = ebuf[(size_t)e * 4 + h] / denom[d * 4 + h];
    const float v = feat_src[(size_t)s * HO + c] * attn;
    atomicAdd(out + (size_t)d * HO + c, v);
}

// ---------------------------------------------------------------------------
extern "C" void kernel_launch(void* const* d_in, const int* in_sizes, int n_in,
                              void* d_out, int out_size, void* d_ws, size_t ws_size,
                              hipStream_t stream)
{
    const float* X    = (const float*)d_in[0];   // [N,256]
    const float* Ein  = (const float*)d_in[1];   // [E,16]
    const int*   src  = (const int*)  d_in[2];   // [E]
    const int*   dst  = (const int*)  d_in[3];   // [E]
    const float* Wsrc = (const float*)d_in[4];   // [128,256]
    const float* Wdst = (const float*)d_in[5];   // [128,256]
    const float* bdst = (const float*)d_in[6];   // [128]
    const float* WaS  = (const float*)d_in[7];   // [4,256]
    const float* WaD  = (const float*)d_in[8];   // [4,256]
    const float* WaE  = (const float*)d_in[9];   // [4,16]

    const int N = in_sizes[0] / DNODE;   // 100000
    const int E = in_sizes[2];           // 1600000
    float* out = (float*)d_out;          // [N,128]

    // workspace layout (256B aligned chunks)
    char* ws = (char*)d_ws;
    size_t off = 0;
    auto alloc = [&](size_t bytes) {
        void* p = ws + off;
        off += (bytes + 255) & ~size_t(255);
        return p;
    };
    float*    feat_src = (float*)   alloc((size_t)N * HO * sizeof(float)); // 51.2MB
    float*    a_src    = (float*)   alloc((size_t)N * 4 * sizeof(float));
    float*    a_dst    = (float*)   alloc((size_t)N * 4 * sizeof(float));
    float*    ebuf     = (float*)   alloc((size_t)E * 4 * sizeof(float));  // 25.6MB
    unsigned* emax_u   = (unsigned*)alloc((size_t)N * 4 * sizeof(unsigned));
    float*    denom    = (float*)   alloc((size_t)N * 4 * sizeof(float));

    // ordered-uint 0 == "below -inf"; denom zeros
    hipMemsetAsync(emax_u, 0, (size_t)N * 4 * sizeof(unsigned), stream);
    hipMemsetAsync(denom,  0, (size_t)N * 4 * sizeof(float),    stream);

    gat_node_proj<<<(N + 15) / 16, 256, 0, stream>>>(
        X, Wsrc, Wdst, bdst, feat_src, out, N);

    gat_node_attn<<<(N + 7) / 8, 256, 0, stream>>>(
        X, WaS, WaD, a_src, a_dst, N);

    gat_edge_logits<<<(E + 255) / 256, 256, 0, stream>>>(
        Ein, src, dst, WaE, a_src, a_dst, ebuf, emax_u, E);

    gat_softmax_norm<<<(E + 255) / 256, 256, 0, stream>>>(
        dst, ebuf, emax_u, denom, E);

    const int aggBlocks = (int)(((size_t)E * HO + 255) / 256);
    gat_aggregate<<<aggBlocks, 256, 0, stream>>>(
        src, dst, feat_src, ebuf, denom, out, E);
}